// HybridFCL_65481071400057
// MI455X (gfx1250) — compile-verified
//
#include <hip/hip_runtime.h>
#include <hip/hip_bf16.h>
#include <math.h>

// Quantum circuit collapses analytically (product state, only 1q gates):
//   exp_vals[b, j] = cos(sum_d theta[d*16 + (15-j)]) * cos((x @ enc_w.T + enc_b)[b, 15-j])
//   logits = exp_vals @ cls_w.T + cls_b
// Fused single kernel; encoding GEMM done with V_WMMA_F32_16X16X4_F32.

typedef __attribute__((ext_vector_type(2))) float v2f;
typedef __attribute__((ext_vector_type(8))) float v8f;

#define NQ    16
#define NF    512
#define NC    10
#define NB    256
#define QDEPTH 4

__global__ __launch_bounds__(32)
void qnn_fused_kernel(const float* __restrict__ x,       // (256, 512)
                      const float* __restrict__ enc_w,   // (16, 512)
                      const float* __restrict__ enc_b,   // (16,)
                      const float* __restrict__ theta,   // (64,)
                      const float* __restrict__ cls_w,   // (10, 16)
                      const float* __restrict__ cls_b,   // (10,)
                      float* __restrict__ out)           // (256, 10)
{
    __shared__ float ldsV[16 * 16];   // val[m][q] tile, m = batch row in tile, q = circuit qubit

    const int lane = threadIdx.x;     // 0..31, wave32, EXEC all ones
    const int nq   = lane & 15;       // A: M row index | B: N column index (circuit qubit)
    const int kh   = lane >> 4;       // K half select (lanes 16-31 hold K=2,3 of each 4-block)
    const int m0   = blockIdx.x * 16; // batch tile base

    // A tile rows from x, B tile columns from enc_w (B = enc_w^T so B[:,n] = enc_w[n,:]).
    const v2f* __restrict__ xa = (const v2f*)(x     + (size_t)(m0 + nq) * NF);
    const v2f* __restrict__ wb = (const v2f*)(enc_w + (size_t)nq * NF);

    // enc[m0+M][N] accumulated over K=512 in steps of 4 via f32 WMMA.
    v8f acc = {0.f, 0.f, 0.f, 0.f, 0.f, 0.f, 0.f, 0.f};
    #pragma unroll 8
    for (int k2 = 0; k2 < NF / 2; k2 += 2) {          // k2 indexes float2's; 4 floats per step
        v2f a = xa[k2 + kh];                          // x[M][k0 + 2*kh .. +1]
        v2f b = wb[k2 + kh];                          // enc_w[N][k0 + 2*kh .. +1]
        acc = __builtin_amdgcn_wmma_f32_16x16x4_f32(
            /*neg_a=*/false, a, /*neg_b=*/false, b,
            /*c_mod=*/(short)0, acc, /*reuse_a=*/false, /*reuse_b=*/false);
    }

    // Per-qubit collapsed gate: <Z_q> = cos(sum_d theta[d*16+q]) * cos(encoded[b,q])
    const float ts = theta[nq] + theta[NQ + nq] + theta[2 * NQ + nq] + theta[3 * NQ + nq];
    const float cT = cosf(ts);
    const float bq = enc_b[nq];

    // C layout: VGPR r -> M = r + 8*kh, N = nq
    #pragma unroll
    for (int r = 0; r < 8; ++r) {
        const int m = r + 8 * kh;
        ldsV[m * 16 + nq] = cT * cosf(acc[r] + bq);
    }
    __syncthreads();

    // Classifier: logits[b,c] = sum_j val[b][15-j] * cls_w[c][j] + cls_b[c]
    // 16 rows x 10 classes = 160 outputs, 5 per lane.
    #pragma unroll
    for (int o = lane; o < 16 * NC; o += 32) {
        const int m = o / NC;
        const int c = o % NC;
        float s = cls_b[c];
        #pragma unroll
        for (int j = 0; j < NQ; ++j)
            s += ldsV[m * 16 + (15 - j)] * cls_w[c * NQ + j];
        out[(size_t)(m0 + m) * NC + c] = s;
    }
}

extern "C" void kernel_launch(void* const* d_in, const int* in_sizes, int n_in,
                              void* d_out, int out_size, void* d_ws, size_t ws_size,
                              hipStream_t stream) {
    const float* x     = (const float*)d_in[0];
    const float* enc_w = (const float*)d_in[1];
    const float* enc_b = (const float*)d_in[2];
    const float* theta = (const float*)d_in[3];
    const float* cls_w = (const float*)d_in[4];
    const float* cls_b = (const float*)d_in[5];
    float* out = (float*)d_out;

    // 16 batch tiles of 16 rows; one wave32 workgroup per tile.
    qnn_fused_kernel<<<NB / 16, 32, 0, stream>>>(x, enc_w, enc_b, theta, cls_w, cls_b, out);
}